// DirGATConv_85822036509289
// MI455X (gfx1250) — compile-verified
//
#include <hip/hip_runtime.h>
#include <hip/hip_bf16.h>
#include <math.h>

#define FIN        128
#define HEADS      4
#define CH         16
#define HC         64          // HEADS*CH
#define NEG_SLOPE  0.2f
#define ALPHA      0.5f
#define EPSV       1e-16f

typedef float v2f __attribute__((ext_vector_type(2)));
typedef float v8f __attribute__((ext_vector_type(8)));

static __device__ __forceinline__ float lrelu(float v) {
    return v > 0.0f ? v : NEG_SLOPE * v;
}
// monotone float <-> uint32 ordinal map (for atomicMax-based segment max)
static __device__ __forceinline__ unsigned f2ord(float f) {
    unsigned u = __float_as_uint(f);
    return (u & 0x80000000u) ? ~u : (u | 0x80000000u);
}
static __device__ __forceinline__ float ord2f(unsigned o) {
    unsigned u = (o & 0x80000000u) ? (o & 0x7fffffffu) : ~o;
    return __uint_as_float(u);
}

// ---------------------------------------------------------------------------
// (1) init: m = ord(-inf), denom = 0, agg = 0   (workspace arrives poisoned)
// ---------------------------------------------------------------------------
__global__ void dirgat_init(unsigned* __restrict__ m0, unsigned* __restrict__ m1,
                            float* __restrict__ den0, float* __restrict__ den1,
                            float* __restrict__ agg0, float* __restrict__ agg1,
                            int nh, int nhc) {
    int t = blockIdx.x * blockDim.x + threadIdx.x;
    if (t < nh) {
        unsigned ninf = f2ord(-INFINITY);
        m0[t] = ninf; m1[t] = ninf;
        den0[t] = 0.0f; den1[t] = 0.0f;
    }
    if (t < nhc) { agg0[t] = 0.0f; agg1[t] = 0.0f; }
}

// ---------------------------------------------------------------------------
// (2) h = x @ W  for both directions, fp32 WMMA (16x16x4), one 16x64 tile/wave.
//     W (128x64 = 32 KB) is staged in LDS once per block and shared by all
//     8 waves; B fragments then come from ds_load instead of global.
//     A frag (16x4 f32): lane = M (both halves), VGPR0=K(+2*hi), VGPR1=K+1(+2*hi)
//     B frag (4x16 f32): lane = N, same K striping
//     C/D (16x16 f32):   VGPR r holds M=r (lanes 0-15) and M=r+8 (lanes 16-31)
// ---------------------------------------------------------------------------
__global__ void dirgat_gemm_wmma(const float* __restrict__ x,
                                 const float* __restrict__ W0,
                                 const float* __restrict__ W1,
                                 float* __restrict__ h0,
                                 float* __restrict__ h1,
                                 int n_nodes) {
    __shared__ float Wl[FIN * HC];            // 32 KB of the 320 KB WGP LDS

    const float* __restrict__ W = blockIdx.y ? W1 : W0;
    float* __restrict__ h       = blockIdx.y ? h1 : h0;

    // cooperative, coalesced fill of W into LDS (before any wave can exit)
    for (int idx = threadIdx.x; idx < FIN * HC; idx += blockDim.x)
        Wl[idx] = W[idx];
    __syncthreads();

    const int wave = threadIdx.x >> 5;
    const int lane = threadIdx.x & 31;
    const int tile = blockIdx.x * (blockDim.x >> 5) + wave;
    if (tile * 16 >= n_nodes) return;         // uniform per-wave: EXEC stays all-1
    const int m0 = tile * 16;
    const int hi = lane >> 4;                 // lane-half selects K pair
    const int l  = lane & 15;

    v8f acc0 = {}, acc1 = {}, acc2 = {}, acc3 = {};
    const float* xr = x + (size_t)(m0 + l) * FIN + 2 * hi;   // A: row m, K base
    const float* wr = Wl + (2 * hi) * HC + l;                // B: row k, col l (LDS)

    for (int k = 0; k < FIN; k += 4) {
        v2f a;  a.x  = xr[0];       a.y  = xr[1];
        v2f b0; b0.x = wr[0];       b0.y = wr[HC + 0];
        v2f b1; b1.x = wr[16];      b1.y = wr[HC + 16];
        v2f b2; b2.x = wr[32];      b2.y = wr[HC + 32];
        v2f b3; b3.x = wr[48];      b3.y = wr[HC + 48];
        acc0 = __builtin_amdgcn_wmma_f32_16x16x4_f32(false, a, false, b0, (short)0, acc0, false, false);
        acc1 = __builtin_amdgcn_wmma_f32_16x16x4_f32(false, a, false, b1, (short)0, acc1, false, false);
        acc2 = __builtin_amdgcn_wmma_f32_16x16x4_f32(false, a, false, b2, (short)0, acc2, false, false);
        acc3 = __builtin_amdgcn_wmma_f32_16x16x4_f32(false, a, false, b3, (short)0, acc3, false, false);
        xr += 4;
        wr += 4 * HC;
    }

    float* hr = h + (size_t)(m0 + hi * 8) * HC + l;
#pragma unroll
    for (int r = 0; r < 8; ++r) {
        hr[r * HC +  0] = acc0[r];
        hr[r * HC + 16] = acc1[r];
        hr[r * HC + 32] = acc2[r];
        hr[r * HC + 48] = acc3[r];
    }
}

// ---------------------------------------------------------------------------
// (3) attention score dots: s_src/s_dst per (node, head), both directions
// ---------------------------------------------------------------------------
__global__ void dirgat_scores(const float* __restrict__ h0, const float* __restrict__ h1,
                              const float* __restrict__ as0, const float* __restrict__ ad0,
                              const float* __restrict__ as1, const float* __restrict__ ad1,
                              float* __restrict__ ss0, float* __restrict__ sd0,
                              float* __restrict__ ss1, float* __restrict__ sd1,
                              int nh) {
    int t = blockIdx.x * blockDim.x + threadIdx.x;
    if (t >= nh) return;
    int node = t >> 2, head = t & 3;
    const float* r0 = h0 + (size_t)node * HC + head * CH;
    const float* r1 = h1 + (size_t)node * HC + head * CH;
    float s0 = 0.f, d0 = 0.f, s1 = 0.f, d1 = 0.f;
#pragma unroll
    for (int c = 0; c < CH; ++c) {
        float v0 = r0[c], v1 = r1[c];
        s0 += v0 * as0[head * CH + c];
        d0 += v0 * ad0[head * CH + c];
        s1 += v1 * as1[head * CH + c];
        d1 += v1 * ad1[head * CH + c];
    }
    ss0[t] = s0; sd0[t] = d0; ss1[t] = s1; sd1[t] = d1;
}

// ---------------------------------------------------------------------------
// (4) segment max over destination neighborhoods (incl. self loops)
// ---------------------------------------------------------------------------
__global__ void dirgat_edge_max(const int* __restrict__ ei, int E, int n,
                                const float* __restrict__ ss0, const float* __restrict__ sd0,
                                const float* __restrict__ ss1, const float* __restrict__ sd1,
                                unsigned* __restrict__ m0, unsigned* __restrict__ m1) {
    int i = blockIdx.x * blockDim.x + threadIdx.x;
    if (i >= E + n) return;
    int s, d;
    if (i < E) { s = ei[i]; d = ei[E + i]; } else { s = d = i - E; }
#pragma unroll
    for (int h = 0; h < HEADS; ++h) {
        float e0 = lrelu(ss0[s * HEADS + h] + sd0[d * HEADS + h]);   // s2d
        atomicMax(&m0[d * HEADS + h], f2ord(e0));
        float e1 = lrelu(ss1[d * HEADS + h] + sd1[s * HEADS + h]);   // d2s (swapped)
        atomicMax(&m1[s * HEADS + h], f2ord(e1));
    }
}

// ---------------------------------------------------------------------------
// (5) segment sum of exp(e - max)
// ---------------------------------------------------------------------------
__global__ void dirgat_edge_sum(const int* __restrict__ ei, int E, int n,
                                const float* __restrict__ ss0, const float* __restrict__ sd0,
                                const float* __restrict__ ss1, const float* __restrict__ sd1,
                                const unsigned* __restrict__ m0, const unsigned* __restrict__ m1,
                                float* __restrict__ den0, float* __restrict__ den1) {
    int i = blockIdx.x * blockDim.x + threadIdx.x;
    if (i >= E + n) return;
    int s, d;
    if (i < E) { s = ei[i]; d = ei[E + i]; } else { s = d = i - E; }
#pragma unroll
    for (int h = 0; h < HEADS; ++h) {
        float e0 = lrelu(ss0[s * HEADS + h] + sd0[d * HEADS + h]);
        atomicAdd(&den0[d * HEADS + h], __expf(e0 - ord2f(m0[d * HEADS + h])));
        float e1 = lrelu(ss1[d * HEADS + h] + sd1[s * HEADS + h]);
        atomicAdd(&den1[s * HEADS + h], __expf(e1 - ord2f(m1[s * HEADS + h])));
    }
}

// ---------------------------------------------------------------------------
// (6) message scatter: 64 threads per edge (coalesced over channels)
// ---------------------------------------------------------------------------
__global__ void dirgat_edge_msg(const int* __restrict__ ei, int E, int n,
                                const float* __restrict__ h0, const float* __restrict__ h1,
                                const float* __restrict__ ss0, const float* __restrict__ sd0,
                                const float* __restrict__ ss1, const float* __restrict__ sd1,
                                const unsigned* __restrict__ m0, const unsigned* __restrict__ m1,
                                const float* __restrict__ den0, const float* __restrict__ den1,
                                float* __restrict__ agg0, float* __restrict__ agg1) {
    long long t = (long long)blockIdx.x * blockDim.x + threadIdx.x;
    long long total = (long long)(E + n) * HC;
    if (t >= total) return;
    int i  = (int)(t >> 6);
    int hc = (int)(t & 63);
    int hd = hc >> 4;
    int s, d;
    if (i < E) { s = ei[i]; d = ei[E + i]; } else { s = d = i - E; }

    // s2d: src=s, dst=d
    float e0 = lrelu(ss0[s * HEADS + hd] + sd0[d * HEADS + hd]);
    float a0 = __expf(e0 - ord2f(m0[d * HEADS + hd])) / (den0[d * HEADS + hd] + EPSV);
    atomicAdd(&agg0[(size_t)d * HC + hc], h0[(size_t)s * HC + hc] * a0);

    // d2s: src=d, dst=s
    float e1 = lrelu(ss1[d * HEADS + hd] + sd1[s * HEADS + hd]);
    float a1 = __expf(e1 - ord2f(m1[s * HEADS + hd])) / (den1[s * HEADS + hd] + EPSV);
    atomicAdd(&agg1[(size_t)s * HC + hc], h1[(size_t)d * HC + hc] * a1);
}

// ---------------------------------------------------------------------------
// (7) blend: out = (1-a)*(agg_s2d + b_s2d) + a*(agg_d2s + b_d2s)
// ---------------------------------------------------------------------------
__global__ void dirgat_final(const float* __restrict__ agg0, const float* __restrict__ agg1,
                             const float* __restrict__ b0, const float* __restrict__ b1,
                             float* __restrict__ out, int nhc) {
    int t = blockIdx.x * blockDim.x + threadIdx.x;
    if (t >= nhc) return;
    int hc = t & 63;
    out[t] = (1.0f - ALPHA) * (agg0[t] + b0[hc]) + ALPHA * (agg1[t] + b1[hc]);
}

extern "C" void kernel_launch(void* const* d_in, const int* in_sizes, int n_in,
                              void* d_out, int out_size, void* d_ws, size_t ws_size,
                              hipStream_t stream) {
    const float* x   = (const float*)d_in[0];
    const int*   ei  = (const int*)d_in[1];
    const float* W0  = (const float*)d_in[2];
    const float* as0 = (const float*)d_in[3];
    const float* ad0 = (const float*)d_in[4];
    const float* b0  = (const float*)d_in[5];
    const float* W1  = (const float*)d_in[6];
    const float* as1 = (const float*)d_in[7];
    const float* ad1 = (const float*)d_in[8];
    const float* b1  = (const float*)d_in[9];
    float* out = (float*)d_out;

    const int n   = in_sizes[0] / FIN;   // 50000
    const int E   = in_sizes[1] / 2;     // 1600000
    const int nh  = n * HEADS;
    const int nhc = n * HC;

    // workspace carve-up (all 4-byte typed)
    char* ws = (char*)d_ws;
    float*    h0   = (float*)ws;    ws += (size_t)nhc * 4;
    float*    h1   = (float*)ws;    ws += (size_t)nhc * 4;
    float*    agg0 = (float*)ws;    ws += (size_t)nhc * 4;
    float*    agg1 = (float*)ws;    ws += (size_t)nhc * 4;
    float*    ss0  = (float*)ws;    ws += (size_t)nh * 4;
    float*    sd0  = (float*)ws;    ws += (size_t)nh * 4;
    float*    ss1  = (float*)ws;    ws += (size_t)nh * 4;
    float*    sd1  = (float*)ws;    ws += (size_t)nh * 4;
    float*    den0 = (float*)ws;    ws += (size_t)nh * 4;
    float*    den1 = (float*)ws;    ws += (size_t)nh * 4;
    unsigned* m0   = (unsigned*)ws; ws += (size_t)nh * 4;
    unsigned* m1   = (unsigned*)ws; ws += (size_t)nh * 4;

    const int B = 256;

    // (1) init accumulators
    dirgat_init<<<(nhc + B - 1) / B, B, 0, stream>>>(m0, m1, den0, den1, agg0, agg1, nh, nhc);

    // (2) WMMA GEMM: 8 waves/block, one 16x64 tile each; grid.y = direction
    {
        int tiles = (n + 15) / 16;
        dim3 grid((tiles + 7) / 8, 2);
        dirgat_gemm_wmma<<<grid, B, 0, stream>>>(x, W0, W1, h0, h1, n);
    }

    // (3) scores
    dirgat_scores<<<(nh + B - 1) / B, B, 0, stream>>>(h0, h1, as0, ad0, as1, ad1,
                                                      ss0, sd0, ss1, sd1, nh);

    const int etot = E + n;
    // (4) segment max
    dirgat_edge_max<<<(etot + B - 1) / B, B, 0, stream>>>(ei, E, n, ss0, sd0, ss1, sd1, m0, m1);
    // (5) segment sum
    dirgat_edge_sum<<<(etot + B - 1) / B, B, 0, stream>>>(ei, E, n, ss0, sd0, ss1, sd1,
                                                          m0, m1, den0, den1);
    // (6) message scatter
    {
        long long total = (long long)etot * HC;
        long long blocks = (total + B - 1) / B;
        dirgat_edge_msg<<<(unsigned)blocks, B, 0, stream>>>(ei, E, n, h0, h1,
                                                            ss0, sd0, ss1, sd1,
                                                            m0, m1, den0, den1, agg0, agg1);
    }
    // (7) blend + bias
    dirgat_final<<<(nhc + B - 1) / B, B, 0, stream>>>(agg0, agg1, b0, b1, out, nhc);
}